// MultiQueryAttention_21732534518167
// MI455X (gfx1250) — compile-verified
//
#include <hip/hip_runtime.h>

#define D_MODEL 1024
#define HEAD_DIM 64
#define BATCH 8
#define SEQ 4096

typedef __attribute__((ext_vector_type(2))) float v2f;
typedef __attribute__((ext_vector_type(8))) float v8f;

// ---------------------------------------------------------------------------
// Kernel 0: V[b][n] = sum_k value[b][k] * Wv[k][n] + bv[n]   (b<8, n<64)
// Tiny (512 dots of length 1024) — one thread per output element.
// ---------------------------------------------------------------------------
__global__ void mqa_vproj(const float* __restrict__ value,
                          const float* __restrict__ Wv,
                          const float* __restrict__ bv,
                          float* __restrict__ V) {
  int t = blockIdx.x * blockDim.x + threadIdx.x;   // 0..511
  if (t >= BATCH * HEAD_DIM) return;
  int b = t >> 6;          // t / 64
  int n = t & 63;          // t % 64
  const float* vp = value + (size_t)b * D_MODEL;
  float acc = bv[n];
  #pragma unroll 4
  for (int k = 0; k < D_MODEL; ++k) {
    acc += vp[k] * Wv[(size_t)k * HEAD_DIM + n];
  }
  V[t] = acc;
}

// ---------------------------------------------------------------------------
// Kernel 1: out_row[m][n] = sum_k tiledV[m][k] * Wo[k][n] + bo[n]
// tiledV[m][k] = V[m][k % 64] for m < 8, 0 for m in [8,16) (zero padding).
// One wave (32 lanes) per 16-wide N tile; K looped in steps of 4 through
// V_WMMA_F32_16X16X4_F32 (full f32 precision — bit-compatible with the
// f32 reference path).
// ---------------------------------------------------------------------------
__global__ void mqa_rowgemm(const float* __restrict__ V,     // [8][64]
                            const float* __restrict__ Wo,    // [1024][1024]
                            const float* __restrict__ bo,    // [1024]
                            float* __restrict__ out_row) {   // [8][1024]
  __shared__ float ldsA[16 * HEAD_DIM];   // 16 M-rows x 64 (period of tiledV in K)

  int lane = threadIdx.x;                 // wave32: 0..31
  // Rows 0-7 = V, rows 8-15 = zero padding for the M=16 WMMA tile.
  for (int i = lane; i < 8 * HEAD_DIM; i += 32) ldsA[i] = V[i];
  for (int i = lane; i < 8 * HEAD_DIM; i += 32) ldsA[8 * HEAD_DIM + i] = 0.0f;
  __syncthreads();

  const int ntile = blockIdx.x;           // 0..63
  const int half  = lane >> 4;            // 0: K+{0,1}, 1: K+{2,3}
  const int l     = lane & 15;
  const int m     = l;                    // A-matrix row held by this lane
  const int n     = ntile * 16 + l;       // B/D column held by this lane

  v8f c = {};
  for (int k = 0; k < D_MODEL; k += 4) {
    const int ka = k + 2 * half;
    v2f a, b;
    // A fragment: lanes 0-15 hold K=k..k+1, lanes 16-31 hold K=k+2..k+3
    a.x = ldsA[m * HEAD_DIM + ( ka      & 63)];
    a.y = ldsA[m * HEAD_DIM + ((ka + 1) & 63)];
    // B fragment: same K split, column n
    b.x = Wo[(size_t)(ka    ) * D_MODEL + n];
    b.y = Wo[(size_t)(ka + 1) * D_MODEL + n];
    // 8 args: (neg_a, A, neg_b, B, c_mod, C, reuse_a, reuse_b)
    c = __builtin_amdgcn_wmma_f32_16x16x4_f32(false, a, false, b,
                                              (short)0, c, false, false);
  }

  const float bias = bo[n];
  if (half == 0) {
    // Lanes 0-15 hold D rows M=0..7 in c[0..7] — exactly the valid batches.
    #pragma unroll
    for (int r = 0; r < 8; ++r)
      out_row[(size_t)r * D_MODEL + n] = c[r] + bias;
  }
}

// ---------------------------------------------------------------------------
// Kernel 2: out[b][s][:] = out_row[b][:] for every s.  134 MB of f32 stores —
// the entire roofline of this problem (~5.8 us at 23.3 TB/s). 128-bit
// coalesced stores, one float4 per thread.
// ---------------------------------------------------------------------------
__global__ void mqa_bcast(const float* __restrict__ out_row,
                          float* __restrict__ out) {
  const size_t row4 = D_MODEL / 4;                       // 256 float4 per row
  size_t i = (size_t)blockIdx.x * blockDim.x + threadIdx.x;
  size_t b = i / ((size_t)SEQ * row4);
  size_t c = i & (row4 - 1);
  float4 v = ((const float4*)out_row)[b * row4 + c];
  ((float4*)out)[i] = v;
}

// ---------------------------------------------------------------------------
extern "C" void kernel_launch(void* const* d_in, const int* in_sizes, int n_in,
                              void* d_out, int out_size, void* d_ws, size_t ws_size,
                              hipStream_t stream) {
  (void)in_sizes; (void)n_in; (void)out_size; (void)ws_size;
  // setup_inputs order:
  // 0:query 1:key 2:value 3:Wq 4:bq 5:Wk 6:bk 7:Wv 8:bv 9:Wo 10:bo
  const float* value = (const float*)d_in[2];
  const float* Wv    = (const float*)d_in[7];
  const float* bv    = (const float*)d_in[8];
  const float* Wo    = (const float*)d_in[9];
  const float* bo    = (const float*)d_in[10];

  float* V       = (float*)d_ws;            // [8][64]
  float* out_row = V + BATCH * HEAD_DIM;    // [8][1024]
  float* out     = (float*)d_out;           // [8][4096][1024]

  mqa_vproj<<<2, 256, 0, stream>>>(value, Wv, bv, V);
  mqa_rowgemm<<<D_MODEL / 16, 32, 0, stream>>>(V, Wo, bo, out_row);

  const size_t total4 = (size_t)BATCH * SEQ * (D_MODEL / 4);   // 8,388,608
  mqa_bcast<<<(unsigned)(total4 / 256), 256, 0, stream>>>(out_row, out);
}